// CrossSigmoidFocalLoss_44203803411084
// MI455X (gfx1250) — compile-verified
//
#include <hip/hip_runtime.h>
#include <hip/hip_bf16.h>

typedef __attribute__((ext_vector_type(2))) float v2f;
typedef __attribute__((ext_vector_type(8))) float v8f;

#define C_CLASSES 80
#define C4 (C_CLASSES / 4)
#define ALPHA_F 0.25f
#define ONE_MINUS_ALPHA_F 0.75f

// Wave-level (32 lane) f32 sum using V_WMMA_F32_16X16X4_F32 twice.
// Step 1: A[M][0] = s_{lane M} (lanes 0-15), A[M][2] = s_{lane M+16} (lanes 16-31),
//         A[*][1]=A[*][3]=0, B = ones  ->  D[M][N] = s_M + s_{M+16}.
//         Summing D's 8 VGPRs gives per-16-lane-half partial totals.
// Step 2: same trick folds the two halves -> every lane holds the full 32-lane sum.
__device__ __forceinline__ float wave_reduce_f32_wmma(float v) {
    v2f a;    a[0] = v;    a[1] = 0.0f;
    v2f ones; ones[0] = 1.0f; ones[1] = 1.0f;
    v8f cz = {};
    v8f d = __builtin_amdgcn_wmma_f32_16x16x4_f32(
        false, a, false, ones, (short)0, cz, false, false);
    float h = d[0] + d[1] + d[2] + d[3] + d[4] + d[5] + d[6] + d[7];
    v2f a2; a2[0] = h; a2[1] = 0.0f;
    v8f d2 = __builtin_amdgcn_wmma_f32_16x16x4_f32(
        false, a2, false, ones, (short)0, cz, false, false);
    return d2[0];
}

// Per-element focal-loss contribution. 1 exp + 1 log + 1 div per element.
__device__ __forceinline__ float focal_elem(float x, bool oh, float mask) {
    float ax = __builtin_fabsf(x);
    float e  = __expf(-ax);            // e^{-|x|} in (0, 1]
    float l  = __logf(1.0f + e);       // log(1 + e^{-|x|})
    // softplus(x) = l + max(x,0); softplus(-x) = l + max(-x,0)
    float sp = oh ? (l + ((x < 0.0f) ? -x : 0.0f))     // bce positive term
                  : (l + ((x > 0.0f) ?  x : 0.0f));    // bce negative term
    float r  = 1.0f / (1.0f + e);
    float p  = (x >= 0.0f) ? r : (1.0f - r);           // sigmoid(x), exact both signs
    float pt = oh ? (1.0f - p) : p;
    float aw = oh ? ALPHA_F : ONE_MINUS_ALPHA_F;
    return mask * aw * pt * pt * sp;
}

__global__ void __launch_bounds__(256)
focal_main_kernel(const float* __restrict__ pred,
                  const int* __restrict__ targets,
                  const float* __restrict__ weight,
                  float* __restrict__ partial,
                  unsigned int total4) {
    const float4* __restrict__ pred4 = (const float4*)pred;
    unsigned int tid    = blockIdx.x * 256u + threadIdx.x;
    unsigned int stride = gridDim.x * 256u;

    float acc = 0.0f;
    for (unsigned int j = tid; j < total4; j += stride) {
        if (j + stride < total4)
            __builtin_prefetch(&pred4[j + stride], 0, 0);   // global_prefetch_b8
        float4 x4 = pred4[j];
        unsigned int row = j / C4;                 // constant divide -> mul-hi
        unsigned int c0  = (j - row * C4) * 4u;
        float wv = weight[row];
        if (wv <= 0.0f) continue;                  // use_weight binarization
        int  t      = targets[row];
        bool negrow = (t == C_CLASSES);
        unsigned int wbits = (unsigned int)wv;     // exact: weight < 2^16
        float xs0 = x4.x, xs1 = x4.y, xs2 = x4.z, xs3 = x4.w;
        float xs[4] = {xs0, xs1, xs2, xs3};
#pragma unroll
        for (int k = 0; k < 4; ++k) {
            unsigned int c = c0 + (unsigned int)k;
            float mask = 1.0f;
            if (negrow)
                mask = (c < 32u) ? (float)((wbits >> c) & 1u) : 0.0f;
            bool oh = (t == (int)c);
            acc += focal_elem(xs[k], oh, mask);
        }
    }

    // Block reduction: WMMA wave reduce, then 8 wave results via LDS.
    float wsum = wave_reduce_f32_wmma(acc);
    __shared__ float sm[8];
    unsigned int lane = threadIdx.x & 31u;
    unsigned int wid  = threadIdx.x >> 5;
    if (lane == 0) sm[wid] = wsum;
    __syncthreads();
    if (threadIdx.x == 0) {
        float s = 0.0f;
#pragma unroll
        for (int i = 0; i < 8; ++i) s += sm[i];
        partial[blockIdx.x] = s;
    }
}

__global__ void __launch_bounds__(256)
focal_finish_kernel(const float* __restrict__ partial,
                    int nparts,
                    float* __restrict__ out,
                    float inv_count) {
    float acc = 0.0f;
    for (int i = (int)threadIdx.x; i < nparts; i += 256) acc += partial[i];
    float wsum = wave_reduce_f32_wmma(acc);
    __shared__ float sm[8];
    unsigned int lane = threadIdx.x & 31u;
    unsigned int wid  = threadIdx.x >> 5;
    if (lane == 0) sm[wid] = wsum;
    __syncthreads();
    if (threadIdx.x == 0) {
        float s = 0.0f;
#pragma unroll
        for (int i = 0; i < 8; ++i) s += sm[i];
        out[0] = s * inv_count;   // LOSS_WEIGHT == 1.0
    }
}

extern "C" void kernel_launch(void* const* d_in, const int* in_sizes, int n_in,
                              void* d_out, int out_size, void* d_ws, size_t ws_size,
                              hipStream_t stream) {
    const float* pred    = (const float*)d_in[0];
    const int*   targets = (const int*)d_in[1];
    const float* weight  = (const float*)d_in[2];
    float*       out     = (float*)d_out;
    float*       partial = (float*)d_ws;

    int total = in_sizes[0];                       // N * C elements
    unsigned int total4 = (unsigned int)(total / 4);

    int nblocks = 1024;                            // 1024*256 threads -> 20 float4/thread
    if (ws_size < (size_t)nblocks * sizeof(float))
        nblocks = (int)(ws_size / sizeof(float));
    if (nblocks < 1) nblocks = 1;

    focal_main_kernel<<<nblocks, 256, 0, stream>>>(pred, targets, weight, partial, total4);
    focal_finish_kernel<<<1, 256, 0, stream>>>(partial, nblocks, out, 1.0f / (float)total);
}